// Att_Beta_Self_LOSS_90709709291843
// MI455X (gfx1250) — compile-verified
//
#include <hip/hip_runtime.h>
#include <math.h>

// Problem constants from the reference: (B, C, H, W) = (16, 4, 512, 512)
#define EPSF      1e-6f
#define HW        (512 * 512)
#define BATCH     16
#define NCH       4
#define NPLANES   (BATCH * NCH)   // 64 (b,c) planes
#define GRIDX     32              // blocks per plane
#define BLOCK     256             // 8 wave32s
#define NPART     (NPLANES * GRIDX)

typedef __attribute__((ext_vector_type(2))) float v2f;
typedef __attribute__((ext_vector_type(8))) float v8f;

// ---------------------------------------------------------------------------
// Cross-lane wave32 sum using V_WMMA_F32_16X16X4_F32.
// A(16x4): lane m holds A[m][0]=v (VGPR0) and A[m][1]=0 (VGPR1); lanes 16..31
// hold K=2,3.  With B(4x16) = all ones:  D[m][n] = v[m] + v[m+16].
// D layout: lane l (0..15) holds D[0..7][l] in 8 VGPRs, lane l+16 holds
// D[8..15][l].  So sum of my 8 D VGPRs plus partner lane (xor 16) = full sum.
// ---------------------------------------------------------------------------
__device__ __forceinline__ float wave_sum_wmma(float v) {
    v2f a; a[0] = v;    a[1] = 0.0f;
    v2f b; b[0] = 1.0f; b[1] = 1.0f;
    v8f c = {};
    c = __builtin_amdgcn_wmma_f32_16x16x4_f32(
            /*neg_a=*/false, a, /*neg_b=*/false, b,
            /*c_mod=*/(short)0, c, /*reuse_a=*/false, /*reuse_b=*/false);
    float s = c[0] + c[1] + c[2] + c[3] + c[4] + c[5] + c[6] + c[7];
    s += __shfl_xor(s, 16, 32);
    return s;   // full 32-lane sum, replicated in every lane
}

// Block-level sum (valid in thread 0). Must be called by ALL threads.
__device__ __forceinline__ float block_sum(float v) {
    __shared__ float smem[BLOCK / 32];
    float wsum = wave_sum_wmma(v);
    const int lane = threadIdx.x & 31;
    const int wave = threadIdx.x >> 5;
    if (lane == 0) smem[wave] = wsum;
    __syncthreads();
    float total = 0.0f;
    if (threadIdx.x == 0) {
        #pragma unroll
        for (int i = 0; i < BLOCK / 32; ++i) total += smem[i];
    }
    return total;
}

// ---------------------------------------------------------------------------
// Kernel 0: zero the per-plane positive counters in workspace.
// ---------------------------------------------------------------------------
__global__ void zero_ws_kernel(float* __restrict__ counts) {
    if (threadIdx.x < NPLANES) counts[threadIdx.x] = 0.0f;
}

// ---------------------------------------------------------------------------
// Kernel 1: num_pos per (b,c) plane.  Counts are integer-valued floats
// (< 2^24), so float atomics are exact and order-independent.
// ---------------------------------------------------------------------------
__global__ void count_pos_kernel(const int* __restrict__ label,
                                 float* __restrict__ counts) {
    const int plane  = blockIdx.y;
    const int* l     = label + (size_t)plane * HW;
    const int stride = gridDim.x * blockDim.x * 4;

    float cnt = 0.0f;
    for (int i = (blockIdx.x * blockDim.x + threadIdx.x) * 4; i < HW; i += stride) {
        int4 lv = *(const int4*)(l + i);
        cnt += ((float)lv.x > 0.5f) ? 1.0f : 0.0f;
        cnt += ((float)lv.y > 0.5f) ? 1.0f : 0.0f;
        cnt += ((float)lv.z > 0.5f) ? 1.0f : 0.0f;
        cnt += ((float)lv.w > 0.5f) ? 1.0f : 0.0f;
    }
    float bs = block_sum(cnt);
    if (threadIdx.x == 0) atomicAdd(&counts[plane], bs);
}

// ---------------------------------------------------------------------------
// Per-element weighted BCE.
// weight = t*alpha*cw^sqrt(1-pc) + (1-t)*(1-alpha)*cw^sqrt(pc), clipped
// bce    = max(p,0) - p*t + log1p(exp(-|p|))
// ---------------------------------------------------------------------------
__device__ __forceinline__ float elem_pow(float p, int li, float alpha, float l2cw) {
    float t   = (float)li;
    float pc  = fminf(fmaxf(p, EPSF), 1.0f - EPSF);
    float wp  = alpha          * exp2f(sqrtf(1.0f - pc) * l2cw);
    float wn  = (1.0f - alpha) * exp2f(sqrtf(pc)        * l2cw);
    float w   = t * wp + (1.0f - t) * wn;
    w         = fminf(fmaxf(w, EPSF), 1.0e6f);
    float bce = fmaxf(p, 0.0f) - p * t + log1pf(expf(-fabsf(p)));
    return bce * w;
}

__device__ __forceinline__ float elem_nopow(float p, int li, float alpha) {
    // cw == 1 exactly (log2(cw) == 0): cw^x == 1.
    float t   = (float)li;
    float w   = t * alpha + (1.0f - t) * (1.0f - alpha);
    w         = fminf(fmaxf(w, EPSF), 1.0e6f);
    float bce = fmaxf(p, 0.0f) - p * t + log1pf(expf(-fabsf(p)));
    return bce * w;
}

// ---------------------------------------------------------------------------
// Kernel 2: main weighted-BCE sum.  Label reads hit L2 (64 MB << 192 MB L2,
// just streamed by kernel 1).  One f32 partial per block (deterministic).
// ---------------------------------------------------------------------------
__global__ void loss_main_kernel(const float* __restrict__ pred,
                                 const int*   __restrict__ label,
                                 const float* __restrict__ cw,
                                 const float* __restrict__ counts,
                                 float*       __restrict__ partials) {
    const int plane = blockIdx.y;
    const int ch    = plane & (NCH - 1);
    const float* p  = pred  + (size_t)plane * HW;
    const int*   l  = label + (size_t)plane * HW;

    const float num_pos = counts[plane];
    const float alpha   = ((float)HW - num_pos) / ((float)HW + EPSF);
    const float l2cw    = log2f(cw[ch]);
    const int   stride  = gridDim.x * blockDim.x * 4;

    float acc = 0.0f;
    if (l2cw == 0.0f) {   // uniform branch: cw == 1 -> skip pow machinery
        for (int i = (blockIdx.x * blockDim.x + threadIdx.x) * 4; i < HW; i += stride) {
            float4 pv = *(const float4*)(p + i);
            int4   lv = *(const int4*)(l + i);
            acc += elem_nopow(pv.x, lv.x, alpha);
            acc += elem_nopow(pv.y, lv.y, alpha);
            acc += elem_nopow(pv.z, lv.z, alpha);
            acc += elem_nopow(pv.w, lv.w, alpha);
        }
    } else {
        for (int i = (blockIdx.x * blockDim.x + threadIdx.x) * 4; i < HW; i += stride) {
            float4 pv = *(const float4*)(p + i);
            int4   lv = *(const int4*)(l + i);
            acc += elem_pow(pv.x, lv.x, alpha, l2cw);
            acc += elem_pow(pv.y, lv.y, alpha, l2cw);
            acc += elem_pow(pv.z, lv.z, alpha, l2cw);
            acc += elem_pow(pv.w, lv.w, alpha, l2cw);
        }
    }

    float bs = block_sum(acc);
    if (threadIdx.x == 0) partials[blockIdx.y * gridDim.x + blockIdx.x] = bs;
}

// ---------------------------------------------------------------------------
// Kernel 3: reduce the 2048 block partials + add B * sum(1000/cw).
// LOSS_WEIGHT == 1.0 so the three multiplies are identity.
// ---------------------------------------------------------------------------
__global__ void finalize_kernel(const float* __restrict__ partials,
                                const float* __restrict__ cw,
                                float*       __restrict__ out) {
    float v = 0.0f;
    for (int i = threadIdx.x; i < NPART; i += blockDim.x) v += partials[i];
    float total = block_sum(v);
    if (threadIdx.x == 0) {
        float extra = 0.0f;
        #pragma unroll
        for (int c = 0; c < NCH; ++c) extra += 1000.0f / cw[c];
        out[0] = total + (float)BATCH * extra;
    }
}

extern "C" void kernel_launch(void* const* d_in, const int* in_sizes, int n_in,
                              void* d_out, int out_size, void* d_ws, size_t ws_size,
                              hipStream_t stream) {
    const float* pred  = (const float*)d_in[0];   // cls_score  f32 [16,4,512,512]
    const int*   label = (const int*)  d_in[1];   // label      i32 [16,4,512,512]
    const float* cw    = (const float*)d_in[2];   // channel_weights f32 [4]
    float*       out   = (float*)d_out;

    float* counts   = (float*)d_ws;          // [64]   per-plane num_pos
    float* partials = counts + NPLANES;      // [2048] per-block partial sums

    zero_ws_kernel<<<1, 64, 0, stream>>>(counts);

    dim3 grid(GRIDX, NPLANES);
    count_pos_kernel<<<grid, BLOCK, 0, stream>>>(label, counts);
    loss_main_kernel<<<grid, BLOCK, 0, stream>>>(pred, label, cw, counts, partials);
    finalize_kernel<<<1, BLOCK, 0, stream>>>(partials, cw, out);
}